// GSageFeatureExtractor_89429809038178
// MI455X (gfx1250) — compile-verified
//
#include <hip/hip_runtime.h>
#include <hip/hip_bf16.h>

typedef __attribute__((ext_vector_type(2))) float v2f;
typedef __attribute__((ext_vector_type(8))) float v8f;

#define D 128
#define GRAPHS 64

// ---------------- utility kernels ----------------

__global__ void fill_zero_kernel(float* __restrict__ p, int n) {
    int i = blockIdx.x * blockDim.x + threadIdx.x;
    if (i < n) p[i] = 0.0f;
}

__global__ void count_deg_kernel(const int* __restrict__ dst, float* __restrict__ cnt, int E) {
    int e = blockIdx.x * blockDim.x + threadIdx.x;
    if (e < E) atomicAdd(&cnt[dst[e]], 1.0f);
}

// 32 threads per edge, each moves 4 floats (float4 gather + 4 f32 atomics)
__global__ void scatter_edges_kernel(const float* __restrict__ x,
                                     const int* __restrict__ src,
                                     const int* __restrict__ dst,
                                     float* __restrict__ agg, int E) {
    unsigned t = blockIdx.x * blockDim.x + threadIdx.x;
    unsigned e = t >> 5;
    if (e >= (unsigned)E) return;
    unsigned c = (t & 31u) << 2;
    int s = src[e], d = dst[e];
    const float4 v = *(const float4*)(x + (size_t)s * D + c);
    float* p = agg + (size_t)d * D + c;
    atomicAdd(p + 0, v.x);
    atomicAdd(p + 1, v.y);
    atomicAdd(p + 2, v.z);
    atomicAdd(p + 3, v.w);
}

// ---------------- fused SAGE layer: mean@Wl + x@Wr + b, relu, l2norm ----------------
// Block: 256 threads (8 waves). Block handles rows [blockIdx.x*16, +16), all 128 cols.
// Wave w computes the 16x16 output tile at cols [16w, 16w+16) via V_WMMA_F32_16X16X4_F32.

__launch_bounds__(256)
__global__ void sage_layer_kernel(const float* __restrict__ xin,
                                  const float* __restrict__ agg,
                                  const float* __restrict__ cnt,
                                  const float* __restrict__ Wl,
                                  const float* __restrict__ Wr,
                                  const float* __restrict__ bias,
                                  float* __restrict__ out,
                                  int do_relu) {
    __shared__ float ssq[16][8];
    __shared__ float inv_norm[16];

    const int lane = threadIdx.x & 31;
    const int wv   = threadIdx.x >> 5;
    const int half = lane >> 4;     // 0: lanes 0-15, 1: lanes 16-31
    const int lr   = lane & 15;
    const int rowbase = blockIdx.x * 16;
    const int row  = rowbase + lr;  // A-fragment row this lane loads
    const int col  = wv * 16 + lr;  // B/C column this lane owns

    const float invc = 1.0f / fmaxf(cnt[row], 1.0f);
    const float* Am = agg + (size_t)row * D;
    const float* Ax = xin + (size_t)row * D;

    v8f acc = {0.f, 0.f, 0.f, 0.f, 0.f, 0.f, 0.f, 0.f};

    // GEMM 1: mean(agg/cnt) @ Wl
#pragma unroll
    for (int k0 = 0; k0 < D; k0 += 4) {
        const int ka = k0 + 2 * half;   // A layout: lanes0-15 K=k0,k0+1 ; lanes16-31 K=k0+2,k0+3
        v2f a, b;
        a.x = Am[ka] * invc;
        a.y = Am[ka + 1] * invc;
        b.x = Wl[(size_t)ka * D + col];
        b.y = Wl[(size_t)(ka + 1) * D + col];
        acc = __builtin_amdgcn_wmma_f32_16x16x4_f32(false, a, false, b,
                                                    (short)0, acc, false, false);
    }
    // GEMM 2: x @ Wr (accumulate into same C)
#pragma unroll
    for (int k0 = 0; k0 < D; k0 += 4) {
        const int ka = k0 + 2 * half;
        v2f a, b;
        a.x = Ax[ka];
        a.y = Ax[ka + 1];
        b.x = Wr[(size_t)ka * D + col];
        b.y = Wr[(size_t)(ka + 1) * D + col];
        acc = __builtin_amdgcn_wmma_f32_16x16x4_f32(false, a, false, b,
                                                    (short)0, acc, false, false);
    }

    // bias + relu
    const float bb = bias[col];
#pragma unroll
    for (int r = 0; r < 8; ++r) {
        float v = acc[r] + bb;
        if (do_relu) v = fmaxf(v, 0.0f);
        acc[r] = v;
    }

    // per-row sum of squares: VGPR r holds row (r + 8*half), 16 cols across a half-wave
#pragma unroll
    for (int r = 0; r < 8; ++r) {
        float s = acc[r] * acc[r];
        s += __shfl_xor(s, 1, 32);
        s += __shfl_xor(s, 2, 32);
        s += __shfl_xor(s, 4, 32);
        s += __shfl_xor(s, 8, 32);
        if (lr == 0) ssq[r + 8 * half][wv] = s;
    }
    __syncthreads();
    if (threadIdx.x < 16) {
        float s = 0.0f;
#pragma unroll
        for (int w = 0; w < 8; ++w) s += ssq[threadIdx.x][w];
        inv_norm[threadIdx.x] = 1.0f / fmaxf(sqrtf(s), 1e-12f);
    }
    __syncthreads();

#pragma unroll
    for (int r = 0; r < 8; ++r) {
        const float sc = inv_norm[r + 8 * half];
        out[(size_t)(rowbase + r + 8 * half) * D + col] = acc[r] * sc;
    }
}

// ---------------- global mean pool ----------------

__global__ void graph_count_kernel(const int* __restrict__ batch, float* __restrict__ gcnt, int N) {
    int n = blockIdx.x * blockDim.x + threadIdx.x;
    if (n < N) atomicAdd(&gcnt[batch[n]], 1.0f);
}

__global__ void pool_sum_kernel(const float* __restrict__ h, const int* __restrict__ batch,
                                float* __restrict__ pool, int N) {
    unsigned t = blockIdx.x * blockDim.x + threadIdx.x;
    unsigned n = t >> 5;
    if (n >= (unsigned)N) return;
    unsigned c = (t & 31u) << 2;
    int g = batch[n];
    const float4 v = *(const float4*)(h + (size_t)n * D + c);
    float* p = pool + (size_t)g * D + c;
    atomicAdd(p + 0, v.x);
    atomicAdd(p + 1, v.y);
    atomicAdd(p + 2, v.z);
    atomicAdd(p + 3, v.w);
}

__global__ void pool_div_kernel(const float* __restrict__ pool, const float* __restrict__ gcnt,
                                float* __restrict__ out) {
    int i = blockIdx.x * blockDim.x + threadIdx.x;
    if (i < GRAPHS * D) {
        int g = i / D;
        out[i] = pool[i] / fmaxf(gcnt[g], 1.0f);
    }
}

// ---------------- host launcher ----------------

static inline unsigned cdiv_u(long long a, long long b) { return (unsigned)((a + b - 1) / b); }

extern "C" void kernel_launch(void* const* d_in, const int* in_sizes, int n_in,
                              void* d_out, int out_size, void* d_ws, size_t ws_size,
                              hipStream_t stream) {
    const float* x    = (const float*)d_in[0];
    const int*   ei   = (const int*)d_in[1];
    const int*   batch= (const int*)d_in[2];
    const float* Wl1  = (const float*)d_in[3];
    const float* Wr1  = (const float*)d_in[4];
    const float* b1   = (const float*)d_in[5];
    const float* Wl2  = (const float*)d_in[6];
    const float* Wr2  = (const float*)d_in[7];
    const float* b2   = (const float*)d_in[8];
    const float* Wl3  = (const float*)d_in[9];
    const float* Wr3  = (const float*)d_in[10];
    const float* b3   = (const float*)d_in[11];
    float* out = (float*)d_out;

    const int N  = in_sizes[0] / D;   // 50000 (divisible by 16)
    const int E  = in_sizes[1] / 2;   // 800000
    const int ND = N * D;

    float* ws   = (float*)d_ws;
    float* agg  = ws;  ws += (size_t)ND;
    float* h1   = ws;  ws += (size_t)ND;
    float* h2   = ws;  ws += (size_t)ND;
    float* cnt  = ws;  ws += N;
    float* gcnt = ws;  ws += GRAPHS;
    float* pool = ws;  ws += (size_t)GRAPHS * D;

    const int* src = ei;
    const int* dstv = ei + E;

    dim3 blk(256);
    const unsigned scat_blocks = cdiv_u((long long)E * 32, 256);
    const unsigned gemm_blocks = (unsigned)(N / 16);

    // in-degree counts (same for every layer)
    fill_zero_kernel<<<cdiv_u(N, 256), blk, 0, stream>>>(cnt, N);
    count_deg_kernel<<<cdiv_u(E, 256), blk, 0, stream>>>(dstv, cnt, E);

    // layer 1
    fill_zero_kernel<<<cdiv_u(ND, 256), blk, 0, stream>>>(agg, ND);
    scatter_edges_kernel<<<scat_blocks, blk, 0, stream>>>(x, src, dstv, agg, E);
    sage_layer_kernel<<<gemm_blocks, blk, 0, stream>>>(x, agg, cnt, Wl1, Wr1, b1, h1, 1);

    // layer 2
    fill_zero_kernel<<<cdiv_u(ND, 256), blk, 0, stream>>>(agg, ND);
    scatter_edges_kernel<<<scat_blocks, blk, 0, stream>>>(h1, src, dstv, agg, E);
    sage_layer_kernel<<<gemm_blocks, blk, 0, stream>>>(h1, agg, cnt, Wl2, Wr2, b2, h2, 1);

    // layer 3 (no relu, still l2-normalized)
    fill_zero_kernel<<<cdiv_u(ND, 256), blk, 0, stream>>>(agg, ND);
    scatter_edges_kernel<<<scat_blocks, blk, 0, stream>>>(h2, src, dstv, agg, E);
    sage_layer_kernel<<<gemm_blocks, blk, 0, stream>>>(h2, agg, cnt, Wl3, Wr3, b3, h1, 0);

    // global mean pool
    fill_zero_kernel<<<1, 64, 0, stream>>>(gcnt, GRAPHS);
    fill_zero_kernel<<<cdiv_u(GRAPHS * D, 256), blk, 0, stream>>>(pool, GRAPHS * D);
    graph_count_kernel<<<cdiv_u(N, 256), blk, 0, stream>>>(batch, gcnt, N);
    pool_sum_kernel<<<cdiv_u((long long)N * 32, 256), blk, 0, stream>>>(h1, batch, pool, N);
    pool_div_kernel<<<cdiv_u(GRAPHS * D, 256), blk, 0, stream>>>(pool, gcnt, out);
}